// RNN_44461501448321
// MI455X (gfx1250) — compile-verified
//
#include <hip/hip_runtime.h>
#include <hip/hip_bf16.h>

// Sizes from the reference
#define BB   128
#define TT   400
#define HH   1024
#define SS   64
#define CC   8
#define N_EXC 819            // int(1024*0.8)
#define ALPHA_N 0.2f         // DT/TAU_N
#define DT_SEC 0.02f
#define AX_SLOW (20.0f/1500.0f)
#define AX_FAST 0.1f
#define U_SLOW 0.45f
#define U_FAST 0.15f

typedef float v2f __attribute__((ext_vector_type(2)));
typedef float v8f __attribute__((ext_vector_type(8)));

__device__ __forceinline__ v8f wmma_f32_k4(v2f a, v2f b, v8f c) {
  // D = A(16x4 f32) * B(4x16 f32) + C(16x16 f32)
  return __builtin_amdgcn_wmma_f32_16x16x4_f32(
      /*neg_a=*/false, a, /*neg_b=*/false, b,
      /*c_mod=*/(short)0, c, /*reuse_a=*/false, /*reuse_b=*/false);
}

// ---------------- prep kernels ----------------

// Effective recurrent weights, K-pair interleaved for b64 B-fragment loads:
// Wp[p][col][r] = ALPHA_N * sign(i) * relu(w_rec[i][col]) * (i!=col), i = 2p+r
__global__ __launch_bounds__(256) void k_prep_W(const float* __restrict__ w_rec,
                                                float* __restrict__ Wp) {
  int idx = blockIdx.x * 256 + threadIdx.x;   // H*H threads
  int i = idx >> 10, j = idx & 1023;
  float v = fmaxf(w_rec[idx], 0.0f);
  if (i == j) v = 0.0f;
  float s = (i < N_EXC) ? 1.0f : -1.0f;
  Wp[(size_t)(i >> 1) * (2 * HH) + j * 2 + (i & 1)] = ALPHA_N * s * v;
}

// K-pair-interleaved transposed drive weights, fused readout P[H][32] + bias[32]
__global__ __launch_bounds__(256) void k_prep_io(
    const float* __restrict__ w_in, const float* __restrict__ w_ctx,
    const float* __restrict__ w_out, const float* __restrict__ b_out,
    const float* __restrict__ c1_w, const float* __restrict__ c1_b,
    const float* __restrict__ c2_w, const float* __restrict__ c2_b,
    float* __restrict__ winP, float* __restrict__ wctxP,
    float* __restrict__ P, float* __restrict__ bias) {
  int h = blockIdx.x * 256 + threadIdx.x;     // H threads
  for (int s = 0; s < SS; ++s)
    winP[(s >> 1) * (2 * HH) + h * 2 + (s & 1)] = ALPHA_N * w_in[h * SS + s];
  for (int c = 0; c < CC; ++c)
    wctxP[(c >> 1) * (2 * HH) + h * 2 + (c & 1)] = ALPHA_N * w_ctx[h * CC + c];
  for (int c = 0; c < 32; ++c) {
    float v = 0.0f;
    if (c < 3) v = w_out[c * HH + h];
    else if (c < 27) {
      int j = (c - 3) >> 3, o = (c - 3) & 7;
      v = c1_w[(j * 2 + 0) * HH + h] * c2_w[(j * 8 + o) * 2 + 0]
        + c1_w[(j * 2 + 1) * HH + h] * c2_w[(j * 8 + o) * 2 + 1];
    }
    P[h * 32 + c] = v;
  }
  if (h == 0) {
    for (int c = 0; c < 32; ++c) {
      float v = 0.0f;
      if (c < 3) v = b_out[c];
      else if (c < 27) {
        int j = (c - 3) >> 3, o = (c - 3) & 7;
        v = c1_b[j * 2 + 0] * c2_w[(j * 8 + o) * 2 + 0]
          + c1_b[j * 2 + 1] * c2_w[(j * 8 + o) * 2 + 1]
          + c2_b[j * 8 + o];
      }
      bias[c] = v;
    }
  }
}

__global__ __launch_bounds__(256) void k_init(float* __restrict__ hbuf,
                                              float* __restrict__ sx,
                                              float* __restrict__ su,
                                              float* __restrict__ hp0) {
  int idx = blockIdx.x * 256 + threadIdx.x;   // B*H threads
  int hc = idx & 1023;
  hbuf[idx] = 0.0f;
  hp0[idx]  = 0.0f;
  sx[idx]   = 1.0f;
  su[idx]   = (hc & 1) ? U_FAST : U_SLOW;
}

// ---------------- per-timestep fused kernel ----------------
// grid = 512 blocks (8 m-tiles x 64 n-tiles), 128 threads = 4 waves.
// Each block computes one 16x16 tile of h_new; K=1024 split across 4 waves.
// Inner loop: one b64 A load + one b64 B load per WMMA.
// Wave 0 also folds in the drive GEMMs (stim K=64, ctx K=8) and runs the
// epilogue (leak+noise+relu, STP update for t+1, writes acts + next h_post).
__global__ __launch_bounds__(128) void k_step(
    const float* __restrict__ Wp, const float* __restrict__ winP,
    const float* __restrict__ wctxP, const float* __restrict__ stim,
    const float* __restrict__ ctx, const float* __restrict__ noise,
    const float* __restrict__ b_rec,
    float* __restrict__ hbuf, float* __restrict__ sx, float* __restrict__ su,
    const float* __restrict__ hp_in, float* __restrict__ hp_out,
    float* __restrict__ acts, int t) {
  __shared__ float red[3][256];

  const int lane = threadIdx.x & 31;
  const int wave = threadIdx.x >> 5;
  const int tile = blockIdx.x;
  const int b0 = (tile & 7) * 16;        // batch tile (M)
  const int h0 = (tile >> 3) * 16;       // unit tile (N)
  const int m  = lane & 15;              // M row (A) / N col (B)
  const int koff = (lane < 16) ? 0 : 2;  // K interleave per ISA layout
  const int phalf = koff >> 1;           // pair offset (0 or 1)

  v8f acc = {};

  // recurrent part: this wave covers K in [wave*256, wave*256+256)
  {
    const float* aP = hp_in + (b0 + m) * HH + wave * 256 + koff;
    const float* bP = Wp + (size_t)(wave * 128 + phalf) * (2 * HH) + (h0 + m) * 2;
    #pragma unroll 8
    for (int k = 0; k < 256; k += 4) {
      v2f a = *(const v2f*)aP;           // h_post: K=koff, koff+1
      v2f b = *(const v2f*)bP;           // W rows  K=koff, koff+1 at this col
      acc = wmma_f32_k4(a, b, acc);
      aP += 4;
      bP += 2 * (2 * HH);                // advance two K-pairs
    }
  }

  if (wave == 0) {                        // wave-uniform branch
    // drive: stimulus (K=64)
    const float* aS = stim + (size_t)(b0 + m) * (TT * SS) + t * SS + koff;
    const float* bS = winP + (size_t)phalf * (2 * HH) + (h0 + m) * 2;
    #pragma unroll 4
    for (int k = 0; k < SS; k += 4) {
      v2f a = *(const v2f*)aS;
      v2f b = *(const v2f*)bS;
      acc = wmma_f32_k4(a, b, acc);
      aS += 4;
      bS += 2 * (2 * HH);
    }
    // drive: context (K=8)
    const float* aC = ctx + (size_t)(b0 + m) * (TT * CC) + t * CC + koff;
    const float* bC = wctxP + (size_t)phalf * (2 * HH) + (h0 + m) * 2;
    #pragma unroll
    for (int k = 0; k < CC; k += 4) {
      v2f a = *(const v2f*)aC;
      v2f b = *(const v2f*)bC;
      acc = wmma_f32_k4(a, b, acc);
      aC += 4;
      bC += 2 * (2 * HH);
    }
  } else {
    #pragma unroll
    for (int r = 0; r < 8; ++r) red[wave - 1][r * 32 + lane] = acc[r];
  }
  __syncthreads();

  if (wave == 0) {
    #pragma unroll
    for (int r = 0; r < 8; ++r)
      acc[r] += red[0][r * 32 + lane] + red[1][r * 32 + lane] + red[2][r * 32 + lane];

    const int hcol = h0 + m;
    const float axv = (hcol & 1) ? AX_FAST : AX_SLOW;
    const float Uv  = (hcol & 1) ? U_FAST : U_SLOW;
    const float brecv = ALPHA_N * b_rec[hcol];

    #pragma unroll
    for (int r = 0; r < 8; ++r) {
      // C/D layout: lanes 0-15 -> M=r, lanes 16-31 -> M=8+r; N = lane%16
      const int b = b0 + r + ((lane < 16) ? 0 : 8);
      const int idx = b * HH + hcol;
      float pre = acc[r] + brecv;                          // a*(d + h_post@W + b_rec)
      float hprev = hbuf[idx];
      float nz = noise[(size_t)t * (BB * HH) + idx];
      float hn = fmaxf(hprev * (1.0f - ALPHA_N) + pre + 0.05f * nz, 0.0f);
      acts[(size_t)b * (TT * HH) + (size_t)t * HH + hcol] = hn;

      // STP update consumed at start of step t+1 (uses h_t)
      float sxv = sx[idx], suv = su[idx];
      float sxn = sxv + axv * (1.0f - sxv) - DT_SEC * suv * sxv * hn;
      sxn = fminf(fmaxf(sxn, 0.0f), 1.0f);
      float sun = suv + axv * (Uv - suv) + DT_SEC * Uv * (1.0f - suv) * hn;
      sun = fminf(fmaxf(sun, 0.0f), 1.0f);
      hbuf[idx] = hn;
      sx[idx] = sxn;
      su[idx] = sun;
      hp_out[idx] = sun * sxn * hn;
    }
  }
}

// ---------------- fused readout: acts[B,T,H] @ P[H,32] + bias ----------------
// one wave per (b,t) row; 8 rows per 256-thread block
__global__ __launch_bounds__(256) void k_project(
    const float* __restrict__ acts, const float* __restrict__ P,
    const float* __restrict__ bias, float* __restrict__ outp,
    float* __restrict__ pcls) {
  const int lane = threadIdx.x & 31;
  const int wave = threadIdx.x >> 5;
  const int row = blockIdx.x * 8 + wave;   // 0 .. B*T-1
  const int b = row / TT;
  const int t = row % TT;
  const float* a = acts + (size_t)b * (TT * HH) + (size_t)t * HH;

  float acc[28];
  #pragma unroll
  for (int c = 0; c < 28; ++c) acc[c] = 0.0f;

  for (int i = 0; i < HH / 32; ++i) {
    int hh = lane + 32 * i;
    float av = a[hh];
    const float4* pr = (const float4*)(P + hh * 32);
    #pragma unroll
    for (int q = 0; q < 7; ++q) {
      float4 pv = pr[q];
      acc[q * 4 + 0] = fmaf(av, pv.x, acc[q * 4 + 0]);
      acc[q * 4 + 1] = fmaf(av, pv.y, acc[q * 4 + 1]);
      acc[q * 4 + 2] = fmaf(av, pv.z, acc[q * 4 + 2]);
      acc[q * 4 + 3] = fmaf(av, pv.w, acc[q * 4 + 3]);
    }
  }

  #pragma unroll
  for (int c = 0; c < 27; ++c) {
    float v = acc[c];
    v += __shfl_xor(v, 16, 32);
    v += __shfl_xor(v, 8, 32);
    v += __shfl_xor(v, 4, 32);
    v += __shfl_xor(v, 2, 32);
    v += __shfl_xor(v, 1, 32);
    if (lane == 0) {
      v += bias[c];
      if (c < 3) outp[((size_t)b * TT + t) * 3 + c] = v;
      else       pcls[((size_t)b * TT + t) * 24 + (c - 3)] = v;
    }
  }
}

// ---------------- launch ----------------
extern "C" void kernel_launch(void* const* d_in, const int* in_sizes, int n_in,
                              void* d_out, int out_size, void* d_ws, size_t ws_size,
                              hipStream_t stream) {
  const float* stim  = (const float*)d_in[0];
  const float* ctx   = (const float*)d_in[1];
  const float* w_rec = (const float*)d_in[2];
  const float* b_rec = (const float*)d_in[3];
  const float* w_in  = (const float*)d_in[4];
  const float* w_ctx = (const float*)d_in[5];
  const float* w_out = (const float*)d_in[6];
  const float* b_out = (const float*)d_in[7];
  const float* c1_w  = (const float*)d_in[8];
  const float* c1_b  = (const float*)d_in[9];
  const float* c2_w  = (const float*)d_in[10];
  const float* c2_b  = (const float*)d_in[11];
  const float* noise = (const float*)d_in[12];

  float* out_outputs = (float*)d_out;                               // [B,T,3]
  float* out_acts    = out_outputs + (size_t)BB * TT * 3;           // [B,T,H]
  float* out_p       = out_acts + (size_t)BB * TT * HH;             // [B,T,3,8]

  float* ws    = (float*)d_ws;
  float* Wp    = ws; ws += (size_t)HH * HH;      // pair-interleaved W_s
  float* winP  = ws; ws += (size_t)SS * HH;      // pair-interleaved a*w_in^T
  float* wctxP = ws; ws += (size_t)CC * HH;      // pair-interleaved a*w_ctx^T
  float* P     = ws; ws += (size_t)HH * 32;
  float* bias  = ws; ws += 32;
  float* hbuf  = ws; ws += (size_t)BB * HH;
  float* sxb   = ws; ws += (size_t)BB * HH;
  float* sub   = ws; ws += (size_t)BB * HH;
  float* hp0   = ws; ws += (size_t)BB * HH;
  float* hp1   = ws; ws += (size_t)BB * HH;

  k_prep_W<<<(HH * HH) / 256, 256, 0, stream>>>(w_rec, Wp);
  k_prep_io<<<HH / 256, 256, 0, stream>>>(w_in, w_ctx, w_out, b_out,
                                          c1_w, c1_b, c2_w, c2_b,
                                          winP, wctxP, P, bias);
  k_init<<<(BB * HH) / 256, 256, 0, stream>>>(hbuf, sxb, sub, hp0);

  for (int t = 0; t < TT; ++t) {
    const float* hin = (t & 1) ? hp1 : hp0;
    float* hout      = (t & 1) ? hp0 : hp1;
    k_step<<<512, 128, 0, stream>>>(Wp, winP, wctxP, stim, ctx, noise, b_rec,
                                    hbuf, sxb, sub, hin, hout, out_acts, t);
  }

  k_project<<<(BB * TT) / 8, 256, 0, stream>>>(out_acts, P, bias,
                                               out_outputs, out_p);
}